// ImprovedSupervisor_46359876993730
// MI455X (gfx1250) — compile-verified
//
#include <hip/hip_runtime.h>
#include <math.h>

// GRU: B=256, T=512, D=H=128, 2 layers, reset_after=True, fp32.
// Strategy:
//   phase 1: xp = X @ W + b_in   (big parallel GEMM via v_wmma_f32_16x16x4_f32)
//   phase 2: sequential-in-T recurrence, batch-tiled 16 rows per workgroup,
//            U kept transposed in LDS (CDNA5 320KB/WGP), h double-buffered in LDS.

typedef __attribute__((ext_vector_type(2))) float v2f;
typedef __attribute__((ext_vector_type(8))) float v8f;

#define GB   256
#define GT   512
#define GD   128
#define GH   128
#define G3H  384
#define LDK  132   // padded LDS stride along K: (132*c + k) % 64 = (4c + k) % 64 -> conflict-free

__device__ __forceinline__ float sig_(float x) { return 1.0f / (1.0f + __expf(-x)); }

// ---------------------------------------------------------------------------
// xp[row, 0:384] = X[row, 0:128] @ W[128, 384] + bias[384]
// one wave -> one 16-row M-tile x four 16-col N-tiles (A fragment reused 4x)
// ---------------------------------------------------------------------------
__global__ __launch_bounds__(256) void gru_xproj(
    const float* __restrict__ X, const float* __restrict__ W,
    const float* __restrict__ bias, float* __restrict__ XP, int rows)
{
  const int lane = threadIdx.x & 31;
  const int wave = threadIdx.x >> 5;
  const int wid  = blockIdx.x * 8 + wave;
  const int mt   = wid / 6;       // 16-row tile
  const int ng   = wid % 6;       // group of 4 N-tiles (64 cols)
  if (mt * 16 >= rows) return;    // wave-uniform

  const int m   = lane & 15;
  const int hi  = lane >> 4;
  const int row = mt * 16 + m;
  const int n0  = ng * 64;

  v8f acc[4];
#pragma unroll
  for (int j = 0; j < 4; ++j) {
    float bv = bias[n0 + j * 16 + m];
    acc[j] = (v8f){bv, bv, bv, bv, bv, bv, bv, bv};
  }

  const float* xrow = X + (size_t)row * GD;
#pragma unroll 4
  for (int k0 = 0; k0 < GD; k0 += 4) {
    const int k = k0 + 2 * hi;                       // lane-half picks K pair
    v2f a = *(const v2f*)(xrow + k);                 // A[m][k], A[m][k+1]
#pragma unroll
    for (int j = 0; j < 4; ++j) {
      const int col = n0 + j * 16 + m;
      v2f b; b.x = W[(size_t)k * G3H + col]; b.y = W[(size_t)(k + 1) * G3H + col];
      acc[j] = __builtin_amdgcn_wmma_f32_16x16x4_f32(
          false, a, false, b, (short)0, acc[j], false, false);
    }
  }

#pragma unroll
  for (int j = 0; j < 4; ++j)
#pragma unroll
    for (int v = 0; v < 8; ++v) {
      const int orow = mt * 16 + v + 8 * hi;         // C layout: VGPR v -> M=v / v+8
      XP[(size_t)orow * G3H + n0 + j * 16 + m] = acc[j][v];
    }
}

// ---------------------------------------------------------------------------
// Recurrence: one block = one 16-row batch tile, loops all T steps.
// Wave w owns hidden cols [16w, 16w+16): computes rec tiles for gates z/r/h
// at cols cw, 128+cw, 256+cw -> gate math stays wave-local.
// ---------------------------------------------------------------------------
__global__ __launch_bounds__(256) void gru_rnn(
    const float* __restrict__ XP,   // [B, T, 384] (input proj incl. b_in)
    const float* __restrict__ U,    // [128, 384]
    const float* __restrict__ brec, // [384]
    float* __restrict__ Y,          // [B, T, 128]
    int T)
{
  __shared__ float sUT[G3H * LDK];     // sUT[c*LDK + k] = U[k][c]  (~198 KB)
  __shared__ float sh[2][16 * LDK];    // double-buffered h tile    (~16.5 KB)

  const int tid  = threadIdx.x;
  const int lane = tid & 31;
  const int wave = tid >> 5;
  const int m    = lane & 15;
  const int hi   = lane >> 4;
  const int b0   = blockIdx.x * 16;

  // stage U transposed into LDS (one time)
  for (int idx = tid; idx < GH * G3H; idx += 256) {
    const int k = idx / G3H, c = idx % G3H;
    sUT[c * LDK + k] = U[idx];
  }
  for (int idx = tid; idx < 16 * LDK; idx += 256) { sh[0][idx] = 0.0f; sh[1][idx] = 0.0f; }
  __syncthreads();

  const int cw = wave * 16 + m;                 // this lane's hidden column
  const float bz = brec[cw];
  const float br = brec[GH + cw];
  const float bh = brec[2 * GH + cw];
  const float* uz = &sUT[(size_t)cw * LDK];
  const float* ur = &sUT[(size_t)(GH + cw) * LDK];
  const float* uh = &sUT[(size_t)(2 * GH + cw) * LDK];

  v8f hreg = (v8f){0, 0, 0, 0, 0, 0, 0, 0};
  int cur = 0;

#pragma unroll 1
  for (int t = 0; t < T; ++t) {
    v8f cz = (v8f){bz, bz, bz, bz, bz, bz, bz, bz};
    v8f cr = (v8f){br, br, br, br, br, br, br, br};
    v8f ch = (v8f){bh, bh, bh, bh, bh, bh, bh, bh};

    const float* ha = &sh[cur][m * LDK];
#pragma unroll 4
    for (int k0 = 0; k0 < GH; k0 += 4) {
      const int k = k0 + 2 * hi;
      v2f a  = *(const v2f*)(ha + k);            // h[m][k], h[m][k+1]
      v2f vz = *(const v2f*)(uz + k);            // U[k][cz], U[k+1][cz]
      v2f vr = *(const v2f*)(ur + k);
      v2f vh = *(const v2f*)(uh + k);
      cz = __builtin_amdgcn_wmma_f32_16x16x4_f32(false, a, false, vz, (short)0, cz, false, false);
      cr = __builtin_amdgcn_wmma_f32_16x16x4_f32(false, a, false, vr, (short)0, cr, false, false);
      ch = __builtin_amdgcn_wmma_f32_16x16x4_f32(false, a, false, vh, (short)0, ch, false, false);
    }

#pragma unroll
    for (int v = 0; v < 8; ++v) {
      const int r_ = v + 8 * hi;                 // C layout row
      const size_t base = ((size_t)(b0 + r_) * T + t) * G3H;
      const float xz = XP[base + cw];
      const float xr = XP[base + GH + cw];
      const float xh = XP[base + 2 * GH + cw];
      const float z  = sig_(xz + cz[v]);
      const float rg = sig_(xr + cr[v]);
      const float hh = tanhf(xh + rg * ch[v]);   // reset applied after recurrent matmul
      const float hn = z * hreg[v] + (1.0f - z) * hh;
      hreg[v] = hn;
      Y[((size_t)(b0 + r_) * T + t) * GH + cw] = hn;
      sh[cur ^ 1][r_ * LDK + cw] = hn;
    }
    __syncthreads();
    cur ^= 1;
  }
}

// ---------------------------------------------------------------------------
extern "C" void kernel_launch(void* const* d_in, const int* in_sizes, int n_in,
                              void* d_out, int out_size, void* d_ws, size_t ws_size,
                              hipStream_t stream)
{
  const float* x  = (const float*)d_in[0];
  const float* W0 = (const float*)d_in[1];
  const float* U0 = (const float*)d_in[2];
  const float* b0 = (const float*)d_in[3];
  const float* W1 = (const float*)d_in[4];
  const float* U1 = (const float*)d_in[5];
  const float* b1 = (const float*)d_in[6];
  float* out = (float*)d_out;

  float* xp = (float*)d_ws;                        // [B*T, 384]  (192 MB)
  float* y0 = xp + (size_t)GB * GT * G3H;          // [B*T, 128]  (64 MB)

  const int rows   = GB * GT;                      // 131072
  const int nwaves = (rows / 16) * 6;              // 49152 waves
  dim3 gblocks(nwaves / 8);                        // 256-thread blocks, 8 waves each

  // layer 0
  gru_xproj<<<gblocks, 256, 0, stream>>>(x, W0, b0, xp, rows);
  gru_rnn  <<<dim3(GB / 16), 256, 0, stream>>>(xp, U0, b0 + G3H, y0, GT);
  // layer 1
  gru_xproj<<<gblocks, 256, 0, stream>>>(y0, W1, b1, xp, rows);
  gru_rnn  <<<dim3(GB / 16), 256, 0, stream>>>(xp, U1, b1 + G3H, out, GT);
}